// Quanv3x3LayerClass_36189394436687
// MI455X (gfx1250) — compile-verified
//
#include <hip/hip_runtime.h>

typedef float v2f __attribute__((ext_vector_type(2)));
typedef float v8f __attribute__((ext_vector_type(8)));

#define NQ   9
#define DIM  512            // 2^9 amplitudes
#define PI_F 3.14159265358979f

// ---------------------------------------------------------------------------
// Kernel 1: build per-channel circuit unitary U_c in a WMMA-fragment-shuffled
// layout so kernel 2 can fetch each A fragment lane-pair with one b64 load:
//   Ushuf[c][(j>>2)*2048 + (d>>4)*64 + ((j>>1)&1)*32 + (d&15)*2 + (j&1)]
//     = <d| U_c |j>
// Each wave simulates one basis column at a time in its private LDS buffer.
// Intra-wave LDS ops are in-order (DScnt), so no workgroup barriers needed.
// ---------------------------------------------------------------------------
__global__ __launch_bounds__(256) void build_unitary(const float* __restrict__ qp,
                                                     float* __restrict__ U) {
  __shared__ float buf[8][2][DIM];       // 8 waves x double buffer x 512 = 32KB
  const int c    = blockIdx.x;           // channel 0..2
  const int lane = threadIdx.x & 31;
  const int w    = threadIdx.x >> 5;     // wave id 0..7
  float* __restrict__ Uc = U + (size_t)c * DIM * DIM;

  for (int col = w; col < DIM; col += 8) {
    float* cur = buf[w][0];
    float* alt = buf[w][1];
    // identity column
    for (int u = 0; u < 16; ++u) {
      int a = u * 32 + lane;
      cur[a] = (a == col) ? 1.0f : 0.0f;
    }
    for (int l = 0; l < 3; ++l) {
      // per-qubit RY(theta), qubit q lives at bit position (8-q)
      for (int q = 0; q < NQ; ++q) {
        float th = 0.5f * qp[(l * 3 + c) * 16 + q];
        float ch = cosf(th), sh = sinf(th);
        int p = 8 - q;
        int lowmask = (1 << p) - 1;
        for (int u = 0; u < 8; ++u) {               // 256 pairs / 32 lanes
          int t  = u * 32 + lane;
          int a0 = ((t >> p) << (p + 1)) | (t & lowmask);
          int a1 = a0 | (1 << p);
          float s0 = cur[a0], s1 = cur[a1];
          cur[a0] = ch * s0 - sh * s1;
          cur[a1] = sh * s0 + ch * s1;
        }
      }
      // CNOT ring (q -> q+1 mod 9) is a basis permutation: out[a] = in[src(a)]
      // src = apply CNOTs in reverse order (each is self-inverse).
      for (int u = 0; u < 16; ++u) {
        int a = u * 32 + lane;
        unsigned s = (unsigned)a;
        #pragma unroll
        for (int q = 8; q >= 0; --q) {
          int cb = 8 - q;                 // control bit
          int tb = 8 - ((q + 1) % 9);     // target bit
          s ^= ((s >> cb) & 1u) << tb;
        }
        alt[a] = cur[s];
      }
      float* tmp = cur; cur = alt; alt = tmp;
    }
    // write column in WMMA-shuffled layout
    {
      int stepBase = (col >> 2) * 2048 + ((col >> 1) & 1) * 32 + (col & 1);
      for (int u = 0; u < 16; ++u) {
        int a = u * 32 + lane;
        Uc[stepBase + (a >> 4) * 64 + (a & 15) * 2] = cur[a];
      }
    }
  }
}

// ---------------------------------------------------------------------------
// Kernel 2: per (site-tile of 16, channel):
//   build Psi[512 x 16] in LDS, Y = U_c * Psi via v_wmma_f32_16x16x4_f32,
//   then z_k = sum_d sign(d,k) * Y[d]^2 reduced with LDS float atomics.
// Grid: (512 site tiles, 3 channels), 256 threads = 8 waves.
// ---------------------------------------------------------------------------
__global__ __launch_bounds__(256) void quanv_gemm(const float* __restrict__ x,
                                                  const float* __restrict__ U,
                                                  float* __restrict__ out) {
  __shared__ float psi[DIM * 16];        // [j][n], stride 16 floats (32KB)
  __shared__ float chs[16][NQ];
  __shared__ float shs[16][NQ];
  __shared__ float pp[16][3][8];         // per-site partial products over qubit triples
  __shared__ float zacc[16 * 3];

  const int tile = blockIdx.x;           // 0..511 (16 sites each)
  const int c    = blockIdx.y;           // channel
  const int tid  = threadIdx.x;
  const int lane = tid & 31;
  const int w    = tid >> 5;

  if (tid < 48) zacc[tid] = 0.0f;

  // ---- phase A: angle-encode the 3x3 patch for 16 sites ----
  if (tid < 16 * NQ) {
    int s = tid / NQ, q = tid % NQ;
    int i = q / 3, j = q % 3;
    int site = tile * 16 + s;
    int ww = site & 31, hh = (site >> 5) & 31, bb = site >> 10;
    int hy = hh + i - 1, wx = ww + j - 1;
    float px = 0.0f;
    if (hy >= 0 && hy < 32 && wx >= 0 && wx < 32)
      px = x[((bb * 32 + hy) * 32 + wx) * 3 + c];
    float half = 0.5f * PI_F * px;
    chs[s][q] = cosf(half);
    shs[s][q] = sinf(half);
  }
  __syncthreads();

  // ---- phase B: partial products over qubit triples (16 sites x 3 groups x 8) ----
  for (int e = tid; e < 16 * 24; e += 256) {
    int s = e / 24, r = e % 24;
    int g = r >> 3, v = r & 7;           // v bit2 = qubit 3g, bit1 = 3g+1, bit0 = 3g+2
    float p0 = ((v >> 2) & 1) ? shs[s][3 * g + 0] : chs[s][3 * g + 0];
    float p1 = ((v >> 1) & 1) ? shs[s][3 * g + 1] : chs[s][3 * g + 1];
    float p2 = ((v     ) & 1) ? shs[s][3 * g + 2] : chs[s][3 * g + 2];
    pp[s][g][v] = p0 * p1 * p2;
  }
  __syncthreads();

  // ---- phase C: expand to full 512-amplitude product states ----
  for (int f = tid; f < DIM * 16; f += 256) {
    int j = f >> 4, n = f & 15;
    psi[f] = pp[n][0][(j >> 6) & 7] * pp[n][1][(j >> 3) & 7] * pp[n][2][j & 7];
  }
  __syncthreads();

  // ---- GEMM: Y[512x16] = Uc[512x512] @ Psi[512x16] ----
  // d-tiles: wave w owns dt = 4w .. 4w+3 (16 rows each).
  // A (16x4 f32) layout: lanes 0-15 hold K=0,1 in vgpr .x/.y; lanes 16-31 K=2,3.
  //   Shuffled U: one v2f (b64) per lane per d-tile per k-step, lane-contiguous.
  // B (4x16 f32) layout: vgpr .x = row K=0 (lanes 0-15) / K=2 (lanes 16-31); .y = K=1 / K=3.
  const float* __restrict__ Uc = U + (size_t)c * DIM * DIM;
  const int m  = lane & 15;
  const int hi = lane >> 4;
  const v2f* __restrict__ aBase = (const v2f*)Uc + (size_t)w * 128 + lane; // + mt*32 + step*1024
  const float* bBase = psi + (2 * hi) * 16 + m;                            // + j0*16 ; .y = +16

  v8f acc0 = {}, acc1 = {}, acc2 = {}, acc3 = {};
  for (int j0 = 0; j0 < DIM; j0 += 4) {
    const v2f* pa = aBase + (size_t)(j0 >> 2) * 1024;
    const float* pb = bBase + j0 * 16;
    v2f b;  b.x = pb[0];  b.y = pb[16];
    v2f a0 = pa[0];
    v2f a1 = pa[32];
    v2f a2 = pa[64];
    v2f a3 = pa[96];
    acc0 = __builtin_amdgcn_wmma_f32_16x16x4_f32(false, a0, false, b, (short)0, acc0, false, false);
    acc1 = __builtin_amdgcn_wmma_f32_16x16x4_f32(false, a1, false, b, (short)0, acc1, false, false);
    acc2 = __builtin_amdgcn_wmma_f32_16x16x4_f32(false, a2, false, b, (short)0, acc2, false, false);
    acc3 = __builtin_amdgcn_wmma_f32_16x16x4_f32(false, a3, false, b, (short)0, acc3, false, false);
  }

  // ---- epilogue: sign(d,k) is constant per 16-row d-tile (bits 8:6 of d = bits 4:2 of dt)
  // C/D layout: vgpr r, lane: site n = lane%16, row d = dt*16 + r + 8*(lane>=16).
#define QUANV_EPI(ACC, MT)                                                        \
  {                                                                               \
    float ss = (ACC)[0] * (ACC)[0] + (ACC)[1] * (ACC)[1] + (ACC)[2] * (ACC)[2] +  \
               (ACC)[3] * (ACC)[3] + (ACC)[4] * (ACC)[4] + (ACC)[5] * (ACC)[5] +  \
               (ACC)[6] * (ACC)[6] + (ACC)[7] * (ACC)[7];                         \
    int dt = w * 4 + (MT);                                                        \
    float s0 = 1.0f - 2.0f * (float)((dt >> 4) & 1);                              \
    float s1 = 1.0f - 2.0f * (float)((dt >> 3) & 1);                              \
    float s2 = 1.0f - 2.0f * (float)((dt >> 2) & 1);                              \
    atomicAdd(&zacc[m * 3 + 0], s0 * ss);                                         \
    atomicAdd(&zacc[m * 3 + 1], s1 * ss);                                         \
    atomicAdd(&zacc[m * 3 + 2], s2 * ss);                                         \
  }
  QUANV_EPI(acc0, 0)
  QUANV_EPI(acc1, 1)
  QUANV_EPI(acc2, 2)
  QUANV_EPI(acc3, 3)
#undef QUANV_EPI

  __syncthreads();
  if (tid < 48) {
    int n = tid / 3, k = tid % 3;
    out[(size_t)(tile * 16 + n) * 9 + c * 3 + k] = zacc[tid];
  }
}

// ---------------------------------------------------------------------------
extern "C" void kernel_launch(void* const* d_in, const int* in_sizes, int n_in,
                              void* d_out, int out_size, void* d_ws, size_t ws_size,
                              hipStream_t stream) {
  const float* x  = (const float*)d_in[0];   // [8,32,32,3] f32
  const float* qp = (const float*)d_in[1];   // [3,3,16]    f32
  float* out = (float*)d_out;                // [8,32,32,9] f32
  float* U   = (float*)d_ws;                 // 3 * 512 * 512 floats = 3MB scratch

  build_unitary<<<dim3(3), dim3(256), 0, stream>>>(qp, U);
  quanv_gemm<<<dim3(512, 3), dim3(256), 0, stream>>>(x, U, out);
}